// LDE_2448131359441
// MI455X (gfx1250) — compile-verified
//
#include <hip/hip_runtime.h>
#include <hip/hip_bf16.h>
#include <math.h>

// Problem constants from the reference
#define Bn 8
#define Tn 2048
#define Dn 64
#define Fn 128
#define BT (Bn * Tn)   // 16384

typedef __attribute__((ext_vector_type(2))) float v2f;
typedef __attribute__((ext_vector_type(8))) float v8f;

// D = A(16x4,f32) * B(4x16,f32) + C(16x16,f32)
__device__ __forceinline__ v8f wmma_f32_16x16x4(v2f a, v2f b, v8f c) {
    return __builtin_amdgcn_wmma_f32_16x16x4_f32(
        /*neg_a=*/false, a, /*neg_b=*/false, b,
        /*c_mod=*/(short)0, c, /*reuse_a=*/false, /*reuse_b=*/false);
}

// ---------------------------------------------------------------------------
// Kernel 0a: xx[row] = sum_f x[row,f]^2   (row in [0, B*T))
// ---------------------------------------------------------------------------
__global__ void k_rownorm(const float* __restrict__ x, float* __restrict__ xx) {
    int row = blockIdx.x * blockDim.x + threadIdx.x;
    if (row >= BT) return;
    const float* p = x + (size_t)row * Fn;
    float s = 0.f;
#pragma unroll 4
    for (int k = 0; k < Fn; k += 4) {
        float4 v = *(const float4*)(p + k);
        s += v.x * v.x + v.y * v.y + v.z * v.z + v.w * v.w;
    }
    xx[row] = s;
}

// ---------------------------------------------------------------------------
// Kernel 0b: dd[d] = ||dic[d,:]||^2 ; lsm[d] = log_softmax(wei)[d]
// single block of 64 threads
// ---------------------------------------------------------------------------
__global__ void k_dic_stats(const float* __restrict__ dic, const float* __restrict__ wei,
                            float* __restrict__ dd, float* __restrict__ lsm) {
    __shared__ float red[Dn];
    int d = threadIdx.x;  // 0..63
    const float* p = dic + d * Fn;
    float s = 0.f;
#pragma unroll 4
    for (int k = 0; k < Fn; ++k) { float v = p[k]; s += v * v; }
    dd[d] = s;

    float w = wei[d];
    red[d] = w; __syncthreads();
    for (int st = 32; st > 0; st >>= 1) {
        if (d < st) red[d] = fmaxf(red[d], red[d + st]);
        __syncthreads();
    }
    float M = red[0]; __syncthreads();
    red[d] = expf(w - M); __syncthreads();
    for (int st = 32; st > 0; st >>= 1) {
        if (d < st) red[d] += red[d + st];
        __syncthreads();
    }
    float lse = M + logf(red[0]);
    lsm[d] = w - lse;
}

// ---------------------------------------------------------------------------
// Kernel 1: G = X * dic^T via WMMA; logits[b][d][t] = -sqrt(xx - 2G + dd) * lsm[d]
// grid = BT/16 blocks, 128 threads (4 waves); wave w handles d-tile w*16.
// ---------------------------------------------------------------------------
__global__ void k_gemm1(const float* __restrict__ x, const float* __restrict__ dic,
                        const float* __restrict__ xx, const float* __restrict__ dd,
                        const float* __restrict__ lsm, float* __restrict__ logits) {
    const int lane = threadIdx.x & 31;
    const int wave = threadIdx.x >> 5;   // 0..3
    const int lm = lane & 15;
    const int hi = lane >> 4;
    const int row0 = blockIdx.x * 16;    // global row in [0, B*T)
    const int d0 = wave * 16;

    const float* xrow = x   + (size_t)(row0 + lm) * Fn;  // A-frag base: row = row0+lm
    const float* drow = dic + (size_t)(d0 + lm) * Fn;    // B-frag base: dic row = d0+lm

    v8f acc0 = {}; v8f acc1 = {};
    for (int k0 = 0; k0 < Fn; k0 += 8) {
        v2f a0 = *(const v2f*)(xrow + k0 + 2 * hi);
        v2f b0 = *(const v2f*)(drow + k0 + 2 * hi);
        v2f a1 = *(const v2f*)(xrow + k0 + 4 + 2 * hi);
        v2f b1 = *(const v2f*)(drow + k0 + 4 + 2 * hi);
        acc0 = wmma_f32_16x16x4(a0, b0, acc0);
        acc1 = wmma_f32_16x16x4(a1, b1, acc1);
    }

    const int d = d0 + lm;               // C/D: N = lane%16 -> d
    const float ddv = dd[d];
    const float ls  = lsm[d];
#pragma unroll
    for (int v = 0; v < 8; ++v) {
        int row = row0 + v + 8 * hi;     // C/D: M = vgpr + 8*(lane/16)
        float G = acc0[v] + acc1[v];
        float sq = xx[row] - 2.0f * G + ddv;
        float dis = sqrtf(fmaxf(sq, 0.0f));
        int b = row >> 11;               // / Tn
        int t = row & (Tn - 1);
        logits[((size_t)(b * Dn + d)) * Tn + t] = -dis * ls;
    }
}

// ---------------------------------------------------------------------------
// Kernel 2: per (b,d): softmax over T in place.  grid = B*D blocks, 256 thr.
// ---------------------------------------------------------------------------
__global__ void k_softmax_t(float* __restrict__ wbuf) {
    float* rowp = wbuf + (size_t)blockIdx.x * Tn;
    __shared__ float red[256];
    const int tid = threadIdx.x;

    float m = -3.402823466e38f;
    for (int t = tid; t < Tn; t += 256) m = fmaxf(m, rowp[t]);
    red[tid] = m; __syncthreads();
    for (int s = 128; s > 0; s >>= 1) {
        if (tid < s) red[tid] = fmaxf(red[tid], red[tid + s]);
        __syncthreads();
    }
    const float M = red[0]; __syncthreads();

    float sum = 0.f;
    for (int t = tid; t < Tn; t += 256) sum += expf(rowp[t] - M);
    red[tid] = sum; __syncthreads();
    for (int s = 128; s > 0; s >>= 1) {
        if (tid < s) red[tid] += red[tid + s];
        __syncthreads();
    }
    const float inv = 1.0f / red[0];

    for (int t = tid; t < Tn; t += 256) rowp[t] = expf(rowp[t] - M) * inv;
}

// ---------------------------------------------------------------------------
// Kernel 3: out[b,d,f] = (P^T X)[d,f] - dic[d,f] via WMMA.
// grid = B * (D/16) blocks, 256 threads (8 waves); wave w -> f-tile w*16.
// ---------------------------------------------------------------------------
__global__ void k_gemm2(const float* __restrict__ wbuf, const float* __restrict__ x,
                        const float* __restrict__ dic, float* __restrict__ out) {
    const int lane = threadIdx.x & 31;
    const int wave = threadIdx.x >> 5;   // 0..7
    const int lm = lane & 15;
    const int hi = lane >> 4;
    const int b  = blockIdx.x >> 2;      // D/16 = 4 d-tiles
    const int d0 = (blockIdx.x & 3) * 16;
    const int f0 = wave * 16;

    // A-frag: A[m,k] = P^T[d0+m, t0+k] = w[b][d0+m][t0+k]; lane row = d0+lm
    const float* wrow = wbuf + ((size_t)(b * Dn + d0 + lm)) * Tn;
    // B-frag: B[k,n] = x[b, t0+k, f0+n]; lane column = f0+lm
    const float* xb = x + (size_t)b * Tn * Fn + f0 + lm;

    v8f acc0 = {}; v8f acc1 = {};
    for (int t0 = 0; t0 < Tn; t0 += 8) {
        v2f a0 = *(const v2f*)(wrow + t0 + 2 * hi);
        v2f a1 = *(const v2f*)(wrow + t0 + 4 + 2 * hi);
        v2f b0, b1;
        b0.x = xb[(size_t)(t0 + 2 * hi) * Fn];
        b0.y = xb[(size_t)(t0 + 2 * hi + 1) * Fn];
        b1.x = xb[(size_t)(t0 + 4 + 2 * hi) * Fn];
        b1.y = xb[(size_t)(t0 + 4 + 2 * hi + 1) * Fn];
        acc0 = wmma_f32_16x16x4(a0, b0, acc0);
        acc1 = wmma_f32_16x16x4(a1, b1, acc1);
    }

#pragma unroll
    for (int v = 0; v < 8; ++v) {
        int d = d0 + v + 8 * hi;         // C/D: M -> d, N -> f
        int f = f0 + lm;
        float val = acc0[v] + acc1[v] - dic[d * Fn + f];
        out[(size_t)b * Dn * Fn + d * Fn + f] = val;
    }
}

// ---------------------------------------------------------------------------
extern "C" void kernel_launch(void* const* d_in, const int* in_sizes, int n_in,
                              void* d_out, int out_size, void* d_ws, size_t ws_size,
                              hipStream_t stream) {
    const float* x   = (const float*)d_in[0];   // (B, T, F)
    const float* dic = (const float*)d_in[1];   // (D, F)
    const float* wei = (const float*)d_in[2];   // (D,)
    float* out = (float*)d_out;                 // (B, D*F)

    float* ws     = (float*)d_ws;
    float* logits = ws;                                   // B*D*T floats (4 MB)
    float* xx     = ws + (size_t)Bn * Dn * Tn;            // BT floats
    float* dd     = xx + BT;                              // D floats
    float* lsm    = dd + Dn;                              // D floats

    k_rownorm  <<<BT / 256, 256, 0, stream>>>(x, xx);
    k_dic_stats<<<1, Dn, 0, stream>>>(dic, wei, dd, lsm);
    k_gemm1    <<<BT / 16, 128, 0, stream>>>(x, dic, xx, dd, lsm, logits);
    k_softmax_t<<<Bn * Dn, 256, 0, stream>>>(logits);
    k_gemm2    <<<Bn * (Dn / 16), 256, 0, stream>>>(logits, x, dic, out);
}